// ActiveNeuralSLAM_66657892434391
// MI455X (gfx1250) — compile-verified
//
#include <hip/hip_runtime.h>
#include <math.h>

// Problem constants (from setup_inputs): N=32, C=16, ego 60x60, allo 480x480, res 5cm
#define HH 480
#define WW 480
#define EH 60
#define EW 60
#define NC 16
#define CG 4              // channels staged per TDM load (4*60*60*4B = 57.6 KB LDS)
#define NGRP (NC / CG)
#define X1 210            // allo_w/2 - ego_w/2
#define Z1 240            // allo_h/2

typedef unsigned int v4u __attribute__((ext_vector_type(4)));
typedef int          v8i __attribute__((ext_vector_type(8)));
typedef int          v4i __attribute__((ext_vector_type(4)));
typedef float        v4f __attribute__((ext_vector_type(4)));

__global__ __launch_bounds__(256)
void place_fused_tdm_kernel(const float* __restrict__ ego,
                            const float* __restrict__ xzrs,
                            float* __restrict__ out)
{
    __shared__ float tile[CG * EH * EW];   // 57,600 bytes
    __shared__ int   blk_any;

    const int n    = blockIdx.y;
    const int tid  = threadIdx.x;
    const int base = blockIdx.x * 256;           // block's first pixel (multiple of 256)
    const int p    = base + tid;                 // 0 .. 480*480-1 (exact)
    const int x    = p % WW;
    const int y    = p / WW;

    // ---- per-batch pose (uniform within block) ----
    const float xm = xzrs[n * 3 + 0];
    const float zm = xzrs[n * 3 + 1];
    const float rd = xzrs[n * 3 + 2];
    // translation grid collapses to a pure pixel shift: ix = x + 240*tx
    const float dx = xm * 20.0f - 240.0f;
    const float dy = zm * 20.0f - 240.0f;
    // theta in (-2*pi, 0]: hardware v_sin/v_cos path is accurate, no Payne-Hanek needed
    const float th = -rd * 0.017453292519943295f;
    const float cs = __cosf(th);
    const float sn = __sinf(th);

    // ---- pass-2 (translation) bilinear corners in intermediate image ----
    const float ixf = (float)x + dx;
    const float iyf = (float)y + dy;
    const float ix0 = floorf(ixf);
    const float iy0 = floorf(iyf);
    const float fx  = ixf - ix0;
    const float fy  = iyf - iy0;

    // 4 translation corners x 4 rotation taps = 16 (weight, ego-tile-offset) pairs,
    // computed once and reused across all 16 channels.
    float tapw[16];
    int   tapo[16];
    int needs = 0;

#pragma unroll
    for (int j = 0; j < 4; ++j) {
        const float xc = ix0 + (float)(j & 1);
        const float yc = iy0 + (float)(j >> 1);
        float Wj = ((j & 1) ? fx : (1.0f - fx)) * ((j >> 1) ? fy : (1.0f - fy));
        // zeros-padding of pass 2: corner outside the 480x480 canvas contributes 0
        if (xc < 0.0f || xc > (float)(WW - 1) || yc < 0.0f || yc > (float)(HH - 1))
            Wj = 0.0f;
        // pass-1 (rotation) sample position for this integer intermediate pixel,
        // expressed directly in ego-tile coordinates
        const float u  = xc - 239.5f;
        const float v  = yc - 239.5f;
        const float ex = cs * u - sn * v + 239.5f - (float)X1;
        const float ey = sn * u + cs * v + 239.5f - (float)Z1;
        const float ex0 = floorf(ex);
        const float ey0 = floorf(ey);
        const float gx  = ex - ex0;
        const float gy  = ey - ey0;
        const int exi = (int)ex0;
        const int eyi = (int)ey0;
#pragma unroll
        for (int b = 0; b < 2; ++b) {
#pragma unroll
            for (int a = 0; a < 2; ++a) {
                const int xx = exi + a;
                const int yy = eyi + b;
                float w = Wj * (a ? gx : (1.0f - gx)) * (b ? gy : (1.0f - gy));
                int off = yy * EW + xx;
                if (xx < 0 || xx >= EW || yy < 0 || yy >= EH) { w = 0.0f; off = 0; }
                const int k = j * 4 + b * 2 + a;
                tapw[k] = w;
                tapo[k] = off;
                needs |= (w != 0.0f) ? 1 : 0;
            }
        }
    }

    // ---- block-level early-out: ~98% of the canvas is provably zero ----
    if (tid == 0) blk_any = 0;
    __syncthreads();
    if (needs) blk_any = 1;
    __syncthreads();

    const size_t hw = (size_t)HH * WW;

    if (!blk_any) {
        // Streaming zero-fill = the kernel's bandwidth-dominant path (~462 MB of the
        // 472 MB output). Each block owns 256 contiguous pixels per channel plane, so
        // 64 threads emit b128 NT stores: 4x fewer store instructions, 512 B/wave-store.
        if (tid < 64) {
            float* zp = out + (size_t)n * NC * hw + (size_t)(base + tid * 4);
            const v4f z = {0.0f, 0.0f, 0.0f, 0.0f};
#pragma unroll
            for (int c = 0; c < NC; ++c)
                __builtin_nontemporal_store(z, (v4f*)(zp + (size_t)c * hw));
        }
        return;
    }

    float* outp = out + (size_t)n * NC * hw + (size_t)p;

    // ---- active block: stage ego tiles into LDS via the Tensor Data Mover ----
    const unsigned lds_base = (unsigned)(size_t)&tile[0];   // low 32 bits = LDS byte offset
    const unsigned long long gbase =
        (unsigned long long)(size_t)ego + (unsigned long long)n * NC * EH * EW * 4ull;

    for (int g = 0; g < NGRP; ++g) {
        if (tid == 0) {
            const unsigned long long ga =
                gbase + (unsigned long long)g * CG * EH * EW * 4ull;
            // D# group0: count=1 | lds_addr | global_addr[56:0] | type=2
            v4u g0 = { 1u,
                       lds_base,
                       (unsigned)(ga & 0xFFFFFFFFu),
                       (unsigned)((ga >> 32) & 0x01FFFFFFu) | 0x80000000u };
            // D# group1: data_size=4B; tensor_dim0=60 (stride 60);
            //            tensor_dim1=60 (stride 3600); tile = 60 x 60 x 4
            v8i g1 = { 0x00020000,          // [17:16] data_size = 2 (4 bytes)
                       0x003C0000,          // [63:48] tensor_dim0 = 60
                       0x003C0000,          // [95:80] tensor_dim1 = 60
                       0x003C0000,          // [127:112] tile_dim0 = 60
                       0x0004003C,          // [143:128] tile_dim1=60, [159:144] tile_dim2=4
                       60,                  // [191:160] tensor_dim0_stride = 60
                       (3600 << 16),        // [223:208] tensor_dim1_stride lo16 = 3600
                       0 };                 // tensor_dim1_stride hi = 0
            // D# group2: tensor_dim2 = 4, rest unused; group3: zeros
            v4i g2 = { CG, 0, 0, 0 };
            v4i g3 = { 0, 0, 0, 0 };
            v8i g4 = { 0, 0, 0, 0, 0, 0, 0, 0 };   // extra operand (clang-23 6-arg form)
            __builtin_amdgcn_tensor_load_to_lds(g0, g1, g2, g3, g4, 0);
            __builtin_amdgcn_s_wait_tensorcnt(0);
        }
        __syncthreads();   // LDS tile visible to all 8 waves

#pragma unroll
        for (int cc = 0; cc < CG; ++cc) {
            const float* t = &tile[cc * EH * EW];
            float acc = 0.0f;
#pragma unroll
            for (int k = 0; k < 16; ++k)
                acc = fmaf(tapw[k], t[tapo[k]], acc);
            __builtin_nontemporal_store(acc, outp + (size_t)(g * CG + cc) * hw);
        }
        __syncthreads();   // protect LDS before next TDM group overwrites it
    }
}

extern "C" void kernel_launch(void* const* d_in, const int* in_sizes, int n_in,
                              void* d_out, int out_size, void* d_ws, size_t ws_size,
                              hipStream_t stream) {
    const float* ego  = (const float*)d_in[0];   // (32,16,60,60) fp32
    const float* xzrs = (const float*)d_in[1];   // (32,3) fp32
    float* out = (float*)d_out;                  // (32,16,480,480) fp32

    const int nbatch = in_sizes[1] / 3;          // 32
    dim3 grid((WW * HH) / 256, nbatch, 1);       // 900 x 32 blocks
    dim3 block(256, 1, 1);                       // 8 waves (wave32)
    hipLaunchKernelGGL(place_fused_tdm_kernel, grid, block, 0, stream,
                       ego, xzrs, out);
}